// GCN_65403761983571
// MI455X (gfx1250) — compile-verified
//
#include <hip/hip_runtime.h>
#include <hip/hip_bf16.h>

typedef float v2f __attribute__((ext_vector_type(2)));
typedef float v8f __attribute__((ext_vector_type(8)));

#define IN_DIM  512
#define HID_DIM 128
#define OUT_DIM 64

// ---------------------------------------------------------------------------
// GEMM1: C[M,128] = A[M,512] @ W[512,128]  via V_WMMA_F32_16X16X4_F32
// One 16x16 tile per wave; 8 waves/block cover 128 columns; 16 rows/block.
// A frag (16x4 f32): lanes 0-15 hold K=0,1 ; lanes 16-31 hold K=2,3
// B frag (4x16 f32): same K split, N = lane&15
// C/D: 8 VGPRs, M = vgpr + 8*(lane>=16), N = lane&15
// ---------------------------------------------------------------------------
__global__ void gcn_gemm1(const float* __restrict__ A,
                          const float* __restrict__ W,
                          float* __restrict__ C, int M)
{
  const int lane = threadIdx.x & 31;
  const int wave = threadIdx.x >> 5;        // 0..7 -> column tile
  const int row0 = blockIdx.x << 4;
  if (row0 >= M) return;
  const int col0 = wave << 4;
  const int half = lane >> 4;               // 0: K even pair, 1: K odd pair base+2
  const int l16  = lane & 15;

  const float* arow = A + (size_t)(row0 + l16) * IN_DIM + 2 * half;
  const float* bcol = W + (size_t)(2 * half) * HID_DIM + col0 + l16;

  v8f acc = {};
#pragma unroll 4
  for (int k = 0; k < IN_DIM; k += 4) {
    v2f a, b;
    a.x = arow[k];
    a.y = arow[k + 1];
    b.x = bcol[(size_t)k * HID_DIM];
    b.y = bcol[(size_t)(k + 1) * HID_DIM];
    acc = __builtin_amdgcn_wmma_f32_16x16x4_f32(false, a, false, b,
                                                (short)0, acc, false, false);
  }

  float* cp = C + (size_t)(row0 + half * 8) * HID_DIM + col0 + l16;
#pragma unroll
  for (int i = 0; i < 8; ++i) cp[(size_t)i * HID_DIM] = acc[i];
}

// ---------------------------------------------------------------------------
// GEMM2: C[M,64] = relu(H[M,128]) @ W[128,64] -- ReLU fused into A-frag load.
// 4 waves/block cover 64 columns; 16 rows/block.
// ---------------------------------------------------------------------------
__global__ void gcn_gemm2(const float* __restrict__ H,
                          const float* __restrict__ W,
                          float* __restrict__ C, int M)
{
  const int lane = threadIdx.x & 31;
  const int wave = threadIdx.x >> 5;        // 0..3
  const int row0 = blockIdx.x << 4;
  if (row0 >= M) return;
  const int col0 = wave << 4;
  const int half = lane >> 4;
  const int l16  = lane & 15;

  const float* arow = H + (size_t)(row0 + l16) * HID_DIM + 2 * half;
  const float* bcol = W + (size_t)(2 * half) * OUT_DIM + col0 + l16;

  v8f acc = {};
#pragma unroll 4
  for (int k = 0; k < HID_DIM; k += 4) {
    v2f a, b;
    a.x = fmaxf(arow[k], 0.0f);
    a.y = fmaxf(arow[k + 1], 0.0f);
    b.x = bcol[(size_t)k * OUT_DIM];
    b.y = bcol[(size_t)(k + 1) * OUT_DIM];
    acc = __builtin_amdgcn_wmma_f32_16x16x4_f32(false, a, false, b,
                                                (short)0, acc, false, false);
  }

  float* cp = C + (size_t)(row0 + half * 8) * OUT_DIM + col0 + l16;
#pragma unroll
  for (int i = 0; i < 8; ++i) cp[(size_t)i * OUT_DIM] = acc[i];
}

// ---------------------------------------------------------------------------
// out[i] = bias[i % D]  (D power of two)  -- accumulator init (bias folded in)
// ---------------------------------------------------------------------------
__global__ void init_bias(float* __restrict__ out, const float* __restrict__ bias,
                          int dmask, size_t total)
{
  size_t i = (size_t)blockIdx.x * blockDim.x + threadIdx.x;
  if (i < total) out[i] = bias[i & (size_t)dmask];
}

// ---------------------------------------------------------------------------
// COO SpMM scatter: Out[row[e], d] += w[e] * S[col[e], d]
// One thread per (edge, feature). Feature dim contiguous -> coalesced
// gathers and coalesced f32 atomics; S and Out tiles live in the 192MB L2.
// ---------------------------------------------------------------------------
__global__ void spmm_scatter(const int* __restrict__ rows,
                             const int* __restrict__ cols,
                             const float* __restrict__ ew,
                             const float* __restrict__ S,
                             float* __restrict__ Out,
                             int shift, long long E)
{
  size_t tid = (size_t)blockIdx.x * blockDim.x + threadIdx.x;
  long long e = (long long)(tid >> shift);
  if (e >= E) return;
  const int D = 1 << shift;
  const int d = (int)(tid & (size_t)(D - 1));
  const int r = rows[e];
  const int c = cols[e];
  const float w = ew[e];
  atomicAdd(Out + (size_t)r * D + d, w * S[(size_t)c * D + d]);
}

// ---------------------------------------------------------------------------
extern "C" void kernel_launch(void* const* d_in, const int* in_sizes, int n_in,
                              void* d_out, int out_size, void* d_ws, size_t ws_size,
                              hipStream_t stream)
{
  const float* x  = (const float*)d_in[0];   // [N,512]
  const int*   ei = (const int*)d_in[1];     // [2,E] flat
  const float* ew = (const float*)d_in[2];   // [E]
  const float* W1 = (const float*)d_in[3];   // [512,128]
  const float* b1 = (const float*)d_in[4];   // [128]
  const float* W2 = (const float*)d_in[5];   // [128,64]
  const float* b2 = (const float*)d_in[6];   // [64]
  float* out = (float*)d_out;                // [N,64]

  const int       N = in_sizes[0] / IN_DIM;  // 50000 (divisible by 16)
  const long long E = in_sizes[2];           // 1.6M
  const int* rows = ei;
  const int* cols = ei + E;

  float* support = (float*)d_ws;                       // N*128 (reused as N*64)
  float* h       = support + (size_t)N * HID_DIM;      // N*128

  // 1) h = b1 (broadcast) : SpMM accumulator with bias pre-added
  {
    size_t total = (size_t)N * HID_DIM;
    init_bias<<<(int)((total + 255) / 256), 256, 0, stream>>>(h, b1, HID_DIM - 1, total);
  }
  // 2) support = x @ W1   (WMMA f32)
  gcn_gemm1<<<N / 16, 256, 0, stream>>>(x, W1, support, N);
  // 3) h += A * support   (edge-parallel atomics, D=128)
  {
    size_t total = (size_t)E << 7;
    spmm_scatter<<<(int)((total + 255) / 256), 256, 0, stream>>>(
        rows, cols, ew, support, h, 7, E);
  }
  // 4) support2 = relu(h) @ W2  (WMMA f32, ReLU fused; reuse support buffer)
  gcn_gemm2<<<N / 16, 128, 0, stream>>>(h, W2, support, N);
  // 5) out = b2 (broadcast)
  {
    size_t total = (size_t)N * OUT_DIM;
    init_bias<<<(int)((total + 255) / 256), 256, 0, stream>>>(out, b2, OUT_DIM - 1, total);
  }
  // 6) out += A * support2  (D=64)
  {
    size_t total = (size_t)E << 6;
    spmm_scatter<<<(int)((total + 255) / 256), 256, 0, stream>>>(
        rows, cols, ew, support, out, 6, E);
  }
}